// Net_4114578670456
// MI455X (gfx1250) — compile-verified
//
#include <hip/hip_runtime.h>

#define N1 12000
#define N2 4000
#define N3 1500
#define N4 500
#define E1C 96000
#define E2C 32000
#define E3C 12000
#define KD 5
#define NK 125

typedef float v2f __attribute__((ext_vector_type(2)));
typedef float v8f __attribute__((ext_vector_type(8)));

__device__ __forceinline__ float eluf(float x) { return x > 0.f ? x : (expf(x) - 1.f); }

// monotonic float <-> uint encoding for atomicMax-based segment max
__device__ __forceinline__ unsigned encf(float f) {
    unsigned b = __float_as_uint(f);
    return (b & 0x80000000u) ? ~b : (b | 0x80000000u);
}
__device__ __forceinline__ float decf(unsigned u) {
    unsigned b = (u & 0x80000000u) ? (u & 0x7fffffffu) : ~u;
    return __uint_as_float(b);
}

// ---------------- spline basis + per-kernel bucket counts ----------------
__global__ void k_basis(const float* __restrict__ ps, int E,
                        float* __restrict__ basis8, int* __restrict__ kidx8,
                        int* __restrict__ counts) {
    int e = blockIdx.x * blockDim.x + threadIdx.x;
    if (e >= E) return;
    float fr[3]; int lo[3];
#pragma unroll
    for (int d = 0; d < 3; ++d) {
        float v = ps[e * 3 + d] * (KD - 1);
        float fl = floorf(v);
        lo[d] = (int)fl;
        fr[d] = v - fl;
    }
#pragma unroll
    for (int b = 0; b < 8; ++b) {
        float w = 1.f; int kf = 0; int mul = 1;
#pragma unroll
        for (int d = 0; d < 3; ++d) {
            int bit = (b >> d) & 1;
            int id = lo[d] + bit;
            id = id < 0 ? 0 : (id > KD - 1 ? KD - 1 : id);
            w *= bit ? fr[d] : (1.f - fr[d]);
            kf += id * mul; mul *= KD;
        }
        basis8[e * 8 + b] = w;
        kidx8[e * 8 + b] = kf;
        atomicAdd(&counts[kf], 1);
    }
}

__global__ void k_deg(const int* __restrict__ dst, int E, int* __restrict__ deg) {
    int e = blockIdx.x * blockDim.x + threadIdx.x;
    if (e >= E) return;
    atomicAdd(&deg[dst[e]], 1);
}

// meta layout (ints): [0..124]=counts [128..252]=offsets [256..381]=grpOff [384..508]=cursor [512]=numGroups
// buckets padded to 64 entries (= 4 WMMA tiles = one 128-thread block)
__global__ void k_scan(int* __restrict__ meta) {
    int* counts  = meta;
    int* offsets = meta + 128;
    int* grpOff  = meta + 256;
    int off = 0, goff = 0;
    for (int k = 0; k < NK; ++k) {
        int c = counts[k];
        offsets[k] = off;
        grpOff[k] = goff;
        int p = (c + 63) & ~63;
        off += p; goff += p >> 6;
    }
    grpOff[NK] = goff;
    meta[512] = goff;
}

__global__ void k_fill(const int* __restrict__ kidx8, const float* __restrict__ basis8,
                       const int* __restrict__ offsets, int* __restrict__ cursor,
                       int* __restrict__ eE, float* __restrict__ eB, int Q) {
    int i = blockIdx.x * blockDim.x + threadIdx.x;
    if (i >= Q) return;
    int k = kidx8[i];
    int pos = offsets[k] + atomicAdd(&cursor[k], 1);
    eE[pos] = i >> 3;        // edge id
    eB[pos] = basis8[i];
}

// ---------------- layer-1 conv (in=1): rank-1, VALU ----------------
__global__ void k_conv_in1(const float* __restrict__ x, const int* __restrict__ src,
                           const int* __restrict__ dst,
                           const float* __restrict__ basis8, const int* __restrict__ kidx8,
                           const float* __restrict__ W /*[125,64]*/,
                           float* __restrict__ agg, int E) {
    int e = blockIdx.x * blockDim.x + threadIdx.x;
    if (e >= E) return;
    float xs = x[src[e]];
    int d = dst[e];
    float bas[8]; int kk[8];
#pragma unroll
    for (int b = 0; b < 8; ++b) { bas[b] = basis8[e * 8 + b]; kk[b] = kidx8[e * 8 + b]; }
    for (int o = 0; o < 64; ++o) {
        float acc = 0.f;
#pragma unroll
        for (int b = 0; b < 8; ++b) acc += bas[b] * W[kk[b] * 64 + o];
        atomicAdd(&agg[d * 64 + o], acc * xs);
    }
}

// ---------------- bucketed WMMA conv ----------------
// 128-thread block = 4 waves; block handles 64 entries (4 tiles) of ONE bucket,
// sharing a single LDS copy of W_k. W_k staged in paired-row layout so each
// B fragment is one contiguous 8-byte LDS load.
template <int O>
__global__ void __launch_bounds__(128) k_conv_gemm(
    const float* __restrict__ X, const int* __restrict__ src, const int* __restrict__ dst,
    const int* __restrict__ eE, const float* __restrict__ eB,
    const int* __restrict__ offsets, const int* __restrict__ grpOff,
    const int* __restrict__ numGroups,
    const float* __restrict__ W /*[125,64,O]*/, float* __restrict__ agg) {
    int g = blockIdx.x;
    if (g >= numGroups[0]) return;
    // uniform binary search: bucket k of this 64-entry group
    int lo = 0, hi = NK;
    while (hi - lo > 1) { int mid = (lo + hi) >> 1; if (grpOff[mid] <= g) lo = mid; else hi = mid; }
    int k = lo;
    int base = offsets[k] + (g - grpOff[k]) * 64;

    __shared__ float lds_w[64 * O];          // paired-row layout
    __shared__ float lds_a[4][16][68];
    __shared__ int   s_dst[4][16];
    __shared__ int   s_src[4][16];
    __shared__ int   s_edge[4][16];
    __shared__ float s_bas[4][16];

    int w = threadIdx.x >> 5;   // wave id: which 16-entry tile
    int l = threadIdx.x & 31;   // lane
    int ebase = base + w * 16;

    if (l < 16) {
        int e = eE[ebase + l];
        s_edge[w][l] = e;
        if (e >= 0) { s_src[w][l] = src[e]; s_dst[w][l] = dst[e]; s_bas[w][l] = eB[ebase + l]; }
        else        { s_src[w][l] = 0;      s_dst[w][l] = 0;      s_bas[w][l] = 0.f; }
    }
    __syncthreads();

    // stage basis-scaled gathered rows (A: 16x64 per wave), float4 loads
    for (int idx = l; idx < 16 * 16; idx += 32) {
        int m = idx >> 4, c4 = idx & 15;
        float4 v = ((const float4*)(X + s_src[w][m] * 64))[c4];
        float bsc = s_bas[w][m];
        v.x *= bsc; v.y *= bsc; v.z *= bsc; v.w *= bsc;
        *(float4*)&lds_a[w][m][c4 * 4] = v;
    }
    // stage W_k (shared by all 4 waves) in paired-row layout:
    // elem(i,o) -> lds_w[((i>>1)*O + o)*2 + (i&1)]
    for (int idx = threadIdx.x; idx < (64 * O) / 4; idx += 128) {
        float4 v = ((const float4*)(W + k * (64 * O)))[idx];
        int i = (idx * 4) / O;
        int o = (idx * 4) % O;
        int pb = ((i >> 1) * O) * 2 + (i & 1);
        lds_w[pb + (o + 0) * 2] = v.x;
        lds_w[pb + (o + 1) * 2] = v.y;
        lds_w[pb + (o + 2) * 2] = v.z;
        lds_w[pb + (o + 3) * 2] = v.w;
    }
    __syncthreads();

    v8f acc[O / 16] = {};
    int m  = l & 15;          // A-row / B-col within tile
    int kh = (l >> 4) * 2;    // K-half selector

#pragma unroll
    for (int kk = 0; kk < 64; kk += 4) {
        v2f a = *(const v2f*)&lds_a[w][m][kk + kh];
#pragma unroll
        for (int t = 0; t < O / 16; ++t) {
            int col = t * 16 + m;
            v2f b = *(const v2f*)&lds_w[(((kk + kh) >> 1) * O + col) * 2];
            acc[t] = __builtin_amdgcn_wmma_f32_16x16x4_f32(
                false, a, false, b, (short)0, acc[t], false, false);
        }
    }
    // scatter-add C rows to agg[dst]
    int half = (l >> 4) * 8;
#pragma unroll
    for (int t = 0; t < O / 16; ++t) {
#pragma unroll
        for (int j = 0; j < 8; ++j) {
            int row = j + half;
            if (s_edge[w][row] >= 0)
                atomicAdd(&agg[s_dst[w][row] * O + t * 16 + m], acc[t][j]);
        }
    }
}

// ---------------- conv epilogue: mean + root weight + bias + ELU ----------------
__global__ void k_finalize(const float* __restrict__ agg, const int* __restrict__ deg,
                           const float* __restrict__ X, const float* __restrict__ root,
                           const float* __restrict__ bias, float* __restrict__ out,
                           int N, int I, int O) {
    int i = blockIdx.x * blockDim.x + threadIdx.x;
    if (i >= N * O) return;
    int n = i / O, o = i % O;
    int d = deg[n]; d = d < 1 ? 1 : d;
    float s = agg[i] / (float)d;
    const float* xr = X + n * I;
    float r = 0.f;
    for (int c = 0; c < I; ++c) r += xr[c] * root[c * O + o];
    out[i] = eluf(s + r + bias[o]);
}

// ---------------- voxel max pool ----------------
__global__ void k_pool_scatter(const float* __restrict__ x, const int* __restrict__ cl,
                               unsigned* __restrict__ buf, int Nin, int F) {
    int i = blockIdx.x * blockDim.x + threadIdx.x;
    if (i >= Nin * F) return;
    int n = i / F, o = i % F;
    atomicMax(&buf[cl[n] * F + o], encf(x[i]));
}
__global__ void k_pool_decode(const unsigned* __restrict__ buf, float* __restrict__ out, int M) {
    int i = blockIdx.x * blockDim.x + threadIdx.x;
    if (i >= M) return;
    out[i] = decf(buf[i]);
}

// ---------------- dense node-wise linear (+optional ELU), strided output ----------------
__global__ void k_linear(const float* __restrict__ X, const float* __restrict__ Wm,
                         const float* __restrict__ b, float* __restrict__ out,
                         int N, int I, int O, int ldo, int off, int act) {
    int i = blockIdx.x * blockDim.x + threadIdx.x;
    if (i >= N * O) return;
    int n = i / O, o = i % O;
    float s = b[o];
    const float* xr = X + n * I;
    for (int c = 0; c < I; ++c) s += xr[c] * Wm[c * O + o];
    if (act) s = eluf(s);
    out[n * ldo + off + o] = s;
}

__global__ void k_gather(const float* __restrict__ X, const int* __restrict__ idx,
                         float* __restrict__ out, int N, int F, int ldo, int off) {
    int i = blockIdx.x * blockDim.x + threadIdx.x;
    if (i >= N * F) return;
    int n = i / F, o = i % F;
    out[n * ldo + off + o] = X[idx[n] * F + o];
}

// ---------------- head: fc2 (64->3) + log_softmax ----------------
__global__ void k_head(const float* __restrict__ X, const float* __restrict__ Wm,
                       const float* __restrict__ b, float* __restrict__ out, int N) {
    int n = blockIdx.x * blockDim.x + threadIdx.x;
    if (n >= N) return;
    float lg[3];
#pragma unroll
    for (int c = 0; c < 3; ++c) {
        float s = b[c];
        for (int i = 0; i < 64; ++i) s += X[n * 64 + i] * Wm[i * 3 + c];
        lg[c] = s;
    }
    float m = fmaxf(lg[0], fmaxf(lg[1], lg[2]));
    float z = expf(lg[0] - m) + expf(lg[1] - m) + expf(lg[2] - m);
    float lz = logf(z);
#pragma unroll
    for (int c = 0; c < 3; ++c) out[n * 3 + c] = lg[c] - m - lz;
}

// =======================================================================
struct SetBuf {
    const int* src; const int* dst;
    int E, Q, P, Ndst;
    float* basis; int* kidx; int* eE; float* eB; int* meta; int* deg;
};

static inline int cdiv(int a, int b) { return (a + b - 1) / b; }

static void buildSet(const SetBuf& s, const float* ps, hipStream_t st) {
    hipMemsetAsync(s.meta, 0, 768 * sizeof(int), st);
    hipMemsetAsync(s.deg, 0, (size_t)s.Ndst * sizeof(int), st);
    hipMemsetAsync(s.eE, 0xFF, (size_t)s.P * sizeof(int), st);
    k_basis<<<cdiv(s.E, 256), 256, 0, st>>>(ps, s.E, s.basis, s.kidx, s.meta);
    k_deg<<<cdiv(s.E, 256), 256, 0, st>>>(s.dst, s.E, s.deg);
    k_scan<<<1, 1, 0, st>>>(s.meta);
    k_fill<<<cdiv(s.Q, 256), 256, 0, st>>>(s.kidx, s.basis, s.meta + 128, s.meta + 384,
                                           s.eE, s.eB, s.Q);
}

template <int O>
static void runConv(const SetBuf& s, const float* X, const float* W,
                    const float* root, const float* bias, float* agg, float* out,
                    int Iroot, hipStream_t st) {
    hipMemsetAsync(agg, 0, (size_t)s.Ndst * O * sizeof(float), st);
    k_conv_gemm<O><<<s.P / 64, 128, 0, st>>>(X, s.src, s.dst, s.eE, s.eB,
                                             s.meta + 128, s.meta + 256, s.meta + 512, W, agg);
    k_finalize<<<cdiv(s.Ndst * O, 256), 256, 0, st>>>(agg, s.deg, X, root, bias, out,
                                                      s.Ndst, Iroot, O);
}

extern "C" void kernel_launch(void* const* d_in, const int* in_sizes, int n_in,
                              void* d_out, int out_size, void* d_ws, size_t ws_size,
                              hipStream_t stream) {
    const float* x1  = (const float*)d_in[0];
    const int*   ei1 = (const int*)d_in[1];
    const float* ps1 = (const float*)d_in[2];
    const int*   c1  = (const int*)d_in[3];
    const int*   ei2 = (const int*)d_in[4];
    const float* ps2 = (const float*)d_in[5];
    const int*   c2  = (const int*)d_in[6];
    const int*   ei3 = (const int*)d_in[7];
    const float* ps3 = (const float*)d_in[8];
    const int*   c3  = (const int*)d_in[9];
    const float* W1 = (const float*)d_in[10], *r1 = (const float*)d_in[11], *b1 = (const float*)d_in[12];
    const float* W2 = (const float*)d_in[13], *r2 = (const float*)d_in[14], *b2 = (const float*)d_in[15];
    const float* W4 = (const float*)d_in[16], *r4 = (const float*)d_in[17], *b4 = (const float*)d_in[18];
    const float* W5 = (const float*)d_in[19], *r5 = (const float*)d_in[20], *b5 = (const float*)d_in[21];
    const float* W6 = (const float*)d_in[22], *r6 = (const float*)d_in[23], *b6 = (const float*)d_in[24];
    const float* fc1_w = (const float*)d_in[25], *fc1_b = (const float*)d_in[26];
    const float* fc2_w = (const float*)d_in[27], *fc2_b = (const float*)d_in[28];
    const float* sk1_w = (const float*)d_in[29], *sk1_b = (const float*)d_in[30];
    const float* sk2_w = (const float*)d_in[31], *sk2_b = (const float*)d_in[32];
    const float* sk3_w = (const float*)d_in[33], *sk3_b = (const float*)d_in[34];
    float* out = (float*)d_out;

    // ---- workspace bump allocator ----
    char* wsb = (char*)d_ws;
    size_t cur = 0;
    auto alloc = [&](size_t bytes) -> void* {
        cur = (cur + 255) & ~(size_t)255;
        void* p = wsb + cur;
        cur += bytes;
        return p;
    };

    SetBuf S1, S2, S3;
    S1.src = ei1; S1.dst = ei1 + E1C; S1.E = E1C; S1.Q = 8 * E1C; S1.P = 8 * E1C + 8192; S1.Ndst = N1;
    S2.src = ei2; S2.dst = ei2 + E2C; S2.E = E2C; S2.Q = 8 * E2C; S2.P = 8 * E2C + 8192; S2.Ndst = N2;
    S3.src = ei3; S3.dst = ei3 + E3C; S3.E = E3C; S3.Q = 8 * E3C; S3.P = 8 * E3C + 8192; S3.Ndst = N3;
    for (SetBuf* s : {&S1, &S2, &S3}) {
        s->basis = (float*)alloc((size_t)s->Q * 4);
        s->kidx  = (int*)  alloc((size_t)s->Q * 4);
        s->eE    = (int*)  alloc((size_t)s->P * 4);
        s->eB    = (float*)alloc((size_t)s->P * 4);
        s->meta  = (int*)  alloc(768 * 4);
        s->deg   = (int*)  alloc((size_t)s->Ndst * 4);
    }
    float* h1   = (float*)alloc((size_t)N1 * 64 * 4);
    float* p2   = (float*)alloc((size_t)N2 * 64 * 4);
    float* h2   = (float*)alloc((size_t)N2 * 64 * 4);
    float* p3   = (float*)alloc((size_t)N3 * 64 * 4);
    float* h3   = (float*)alloc((size_t)N3 * 64 * 4);
    float* p4   = (float*)alloc((size_t)N4 * 64 * 4);
    float* h4   = (float*)alloc((size_t)N4 * 32 * 4);
    float* cat3 = (float*)alloc((size_t)N3 * 64 * 4);
    float* h3b  = (float*)alloc((size_t)N3 * 32 * 4);
    float* cat2 = (float*)alloc((size_t)N2 * 64 * 4);
    float* h2b  = (float*)alloc((size_t)N2 * 32 * 4);
    float* cat1 = (float*)alloc((size_t)N1 * 64 * 4);
    float* h1b  = (float*)alloc((size_t)N1 * 64 * 4);
    float* agg  = (float*)alloc((size_t)N1 * 64 * 4);
    unsigned* pool = (unsigned*)alloc((size_t)N2 * 64 * 4);

    // ---- build edge buckets (shared by both convs per edge set) ----
    buildSet(S1, ps1, stream);
    buildSet(S2, ps2, stream);
    buildSet(S3, ps3, stream);

    // ---- conv1 (in=1 special case) ----
    hipMemsetAsync(agg, 0, (size_t)N1 * 64 * 4, stream);
    k_conv_in1<<<cdiv(E1C, 256), 256, 0, stream>>>(x1, S1.src, S1.dst, S1.basis, S1.kidx, W1, agg, E1C);
    k_finalize<<<cdiv(N1 * 64, 256), 256, 0, stream>>>(agg, S1.deg, x1, r1, b1, h1, N1, 1, 64);

    // ---- pool1: N1 -> N2 ----
    hipMemsetAsync(pool, 0, (size_t)N2 * 64 * 4, stream);
    k_pool_scatter<<<cdiv(N1 * 64, 256), 256, 0, stream>>>(h1, c1, pool, N1, 64);
    k_pool_decode<<<cdiv(N2 * 64, 256), 256, 0, stream>>>(pool, p2, N2 * 64);

    // ---- conv2 (WMMA) ----
    runConv<64>(S2, p2, W2, r2, b2, agg, h2, 64, stream);

    // ---- pool2: N2 -> N3 ----
    hipMemsetAsync(pool, 0, (size_t)N3 * 64 * 4, stream);
    k_pool_scatter<<<cdiv(N2 * 64, 256), 256, 0, stream>>>(h2, c2, pool, N2, 64);
    k_pool_decode<<<cdiv(N3 * 64, 256), 256, 0, stream>>>(pool, p3, N3 * 64);

    // ---- conv3 (reuses W2, WMMA) ----
    runConv<64>(S3, p3, W2, r2, b2, agg, h3, 64, stream);

    // ---- pool3: N3 -> N4 ----
    hipMemsetAsync(pool, 0, (size_t)N4 * 64 * 4, stream);
    k_pool_scatter<<<cdiv(N3 * 64, 256), 256, 0, stream>>>(h3, c3, pool, N3, 64);
    k_pool_decode<<<cdiv(N4 * 64, 256), 256, 0, stream>>>(pool, p4, N4 * 64);

    // ---- fc1 + ELU ----
    k_linear<<<cdiv(N4 * 32, 256), 256, 0, stream>>>(p4, fc1_w, fc1_b, h4, N4, 64, 32, 32, 0, 1);

    // ---- cat3 = [h4[c3], h3@sk3+b] ; conv4 (WMMA, O=32) ----
    k_gather<<<cdiv(N3 * 32, 256), 256, 0, stream>>>(h4, c3, cat3, N3, 32, 64, 0);
    k_linear<<<cdiv(N3 * 32, 256), 256, 0, stream>>>(h3, sk3_w, sk3_b, cat3, N3, 64, 32, 64, 32, 0);
    runConv<32>(S3, cat3, W4, r4, b4, agg, h3b, 64, stream);

    // ---- cat2 ; conv5 (WMMA, O=32) ----
    k_gather<<<cdiv(N2 * 32, 256), 256, 0, stream>>>(h3b, c2, cat2, N2, 32, 64, 0);
    k_linear<<<cdiv(N2 * 32, 256), 256, 0, stream>>>(h2, sk2_w, sk2_b, cat2, N2, 64, 32, 64, 32, 0);
    runConv<32>(S2, cat2, W5, r5, b5, agg, h2b, 64, stream);

    // ---- cat1 ; conv6 (WMMA, O=64) ----
    k_gather<<<cdiv(N1 * 32, 256), 256, 0, stream>>>(h2b, c1, cat1, N1, 32, 64, 0);
    k_linear<<<cdiv(N1 * 32, 256), 256, 0, stream>>>(h1, sk1_w, sk1_b, cat1, N1, 64, 32, 64, 32, 0);
    runConv<64>(S1, cat1, W6, r6, b6, agg, h1b, 64, stream);

    // ---- head ----
    k_head<<<cdiv(N1, 256), 256, 0, stream>>>(h1b, fc2_w, fc2_b, out, N1);
    (void)in_sizes; (void)n_in; (void)out_size; (void)ws_size;
}